// Net_81552839016510
// MI455X (gfx1250) — compile-verified
//
#include <hip/hip_runtime.h>
#include <stdint.h>

#define NN   100000    // nodes
#define NE   1600000   // edges
#define FIN  50
#define FINP 64        // padded K for layer 1
#define HID  256
#define NCLS 121
#define NCLSP 128      // padded final width

typedef __attribute__((ext_vector_type(2))) float v2f;
typedef __attribute__((ext_vector_type(8))) float v8f;
typedef __attribute__((ext_vector_type(4))) int   v4i;

#if __has_builtin(__builtin_amdgcn_global_load_async_to_lds_b128) && \
    __has_builtin(__builtin_amdgcn_global_load_async_to_lds_b32)  && \
    __has_builtin(__builtin_amdgcn_s_wait_asynccnt)
#define USE_ASYNC 1
typedef __attribute__((address_space(1))) v4i  g_v4i;
typedef __attribute__((address_space(3))) v4i  l_v4i;
typedef __attribute__((address_space(1))) int  g_i32;
typedef __attribute__((address_space(3))) int  l_i32;
#else
#define USE_ASYNC 0
#endif

// ---------------------------------------------------------------- degrees
__global__ void k_init_deg(float* deg) {
    int i = blockIdx.x * 256 + threadIdx.x;
    if (i < NN) deg[i] = 1.0f;                       // self loop
}

__global__ void k_deg_edges(const long long* __restrict__ ei, float* deg) {
    int e = blockIdx.x * 256 + threadIdx.x;
    if (e < NE) atomicAdd(&deg[(int)ei[NE + e]], 1.0f);
}

__global__ void k_rsqrt_inplace(float* deg) {
    int i = blockIdx.x * 256 + threadIdx.x;
    if (i < NN) deg[i] = rsqrtf(deg[i]);
}

// ---------------------------------------------------------------- padding
__global__ void k_pad_x(const float* __restrict__ x, float* __restrict__ xp) {
    int i = blockIdx.x * 256 + threadIdx.x;
    if (i >= NN * FINP) return;
    int r = i >> 6, c = i & 63;
    float v = x[r * FIN + min(c, FIN - 1)];
    xp[i] = (c < FIN) ? v : 0.0f;
}

__global__ void k_pad_mat(const float* __restrict__ src, float* __restrict__ dst,
                          int R, int C, int Rp, int Cp) {
    int i = blockIdx.x * 256 + threadIdx.x;
    if (i >= Rp * Cp) return;
    int r = i / Cp, c = i % Cp;
    float v = src[min(r, R - 1) * C + min(c, C - 1)];
    dst[i] = (r < R && c < C) ? v : 0.0f;
}

// ---------------------------------------------------------------- GEMM (WMMA f32)
// C[M,NCOUT] = A[M,K] @ B[K,NC].  K%16==0, NC%128==0 (pre-padded), NCOUT<=NC.
// Block: 256 threads / 8 waves, tile 64(M) x 128(N).
// Wave w: wmw=w&1 -> 32-row half, wnw=w>>1 -> 32-col group. 2x2 fragments.
//   EPI 0: C *= dinv[row]; EPI 1: relu(C+bias); EPI 2: C+bias

template <int K, int NC>
__device__ __forceinline__ void stage_tile(
    const float* __restrict__ A, const float* __restrict__ B,
    float* al, float* bt, int tid, int m0, int n0, int kb, int M)
{
    // A tile: 64 rows x 16 k, pitch 20, one b128 per thread (row clamped)
    int row = tid >> 2;
    int seg = (tid & 3) * 4;
    int gr  = min(m0 + row, M - 1);
    const float* gpA = A + (size_t)gr * K + kb + seg;
    float* lpA = al + row * 20 + seg;
    // B tile: 16 k x 128 cols, stored transposed bt[col*20 + k]
    int r  = tid >> 4;
    int cs = (tid & 15) * 8;
    const float* gpB = B + (size_t)(kb + r) * NC + n0 + cs;
#if USE_ASYNC
    __builtin_amdgcn_global_load_async_to_lds_b128((g_v4i*)gpA, (l_v4i*)lpA, 0, 0);
    #pragma unroll
    for (int j = 0; j < 8; ++j)
        __builtin_amdgcn_global_load_async_to_lds_b32((g_i32*)(gpB + j),
                                                      (l_i32*)(bt + (cs + j) * 20 + r), 0, 0);
#else
    *(float4*)lpA = *(const float4*)gpA;
    float4 bv0 = *(const float4*)gpB;
    float4 bv1 = *(const float4*)(gpB + 4);
    bt[(cs + 0) * 20 + r] = bv0.x;
    bt[(cs + 1) * 20 + r] = bv0.y;
    bt[(cs + 2) * 20 + r] = bv0.z;
    bt[(cs + 3) * 20 + r] = bv0.w;
    bt[(cs + 4) * 20 + r] = bv1.x;
    bt[(cs + 5) * 20 + r] = bv1.y;
    bt[(cs + 6) * 20 + r] = bv1.z;
    bt[(cs + 7) * 20 + r] = bv1.w;
#endif
}

template <int EPI, int K, int NC, int NCOUT>
__global__ __launch_bounds__(256) void k_gemm(
    const float* __restrict__ A, const float* __restrict__ B,
    float* __restrict__ C, int M,
    const float* __restrict__ dinv, const float* __restrict__ bias)
{
    __shared__ float Al[2][64 * 20];
    __shared__ float Bt[2][128 * 20];

    const int tid  = threadIdx.x;
    const int lane = tid & 31;
    const int w    = tid >> 5;
    const int wmw  = w & 1;        // 32-row half
    const int wnw  = w >> 1;       // 32-col group (0..3)
    const int r16  = lane & 15;
    const int kh   = lane >> 4;
    const int m0   = blockIdx.x * 64;
    const int n0   = blockIdx.y * 128;

    const int aoff0 = (wmw * 32 + r16) * 20 + 2 * kh;
    const int aoff1 = aoff0 + 16 * 20;
    const int boff0 = (wnw * 32 + r16) * 20 + 2 * kh;
    const int boff1 = boff0 + 16 * 20;

    v8f c00 = {}, c01 = {}, c10 = {}, c11 = {};
    const int nk = K >> 4;

#if USE_ASYNC
    stage_tile<K, NC>(A, B, Al[0], Bt[0], tid, m0, n0, 0, M);
    for (int i = 0; i < nk; ++i) {
        __builtin_amdgcn_s_wait_asynccnt(0);
        __syncthreads();
        if (i + 1 < nk) {
            int nb = (i + 1) & 1;
            stage_tile<K, NC>(A, B, Al[nb], Bt[nb], tid, m0, n0, (i + 1) << 4, M);
        }
        const float* al = Al[i & 1];
        const float* bt = Bt[i & 1];
        #pragma unroll
        for (int kk = 0; kk < 16; kk += 4) {
            v2f a0 = *(const v2f*)&al[aoff0 + kk];
            v2f a1 = *(const v2f*)&al[aoff1 + kk];
            v2f b0 = *(const v2f*)&bt[boff0 + kk];
            v2f b1 = *(const v2f*)&bt[boff1 + kk];
            c00 = __builtin_amdgcn_wmma_f32_16x16x4_f32(false, a0, false, b0, (short)0, c00, false, false);
            c01 = __builtin_amdgcn_wmma_f32_16x16x4_f32(false, a0, false, b1, (short)0, c01, false, false);
            c10 = __builtin_amdgcn_wmma_f32_16x16x4_f32(false, a1, false, b0, (short)0, c10, false, false);
            c11 = __builtin_amdgcn_wmma_f32_16x16x4_f32(false, a1, false, b1, (short)0, c11, false, false);
        }
    }
#else
    for (int i = 0; i < nk; ++i) {
        stage_tile<K, NC>(A, B, Al[0], Bt[0], tid, m0, n0, i << 4, M);
        __syncthreads();
        const float* al = Al[0];
        const float* bt = Bt[0];
        #pragma unroll
        for (int kk = 0; kk < 16; kk += 4) {
            v2f a0 = *(const v2f*)&al[aoff0 + kk];
            v2f a1 = *(const v2f*)&al[aoff1 + kk];
            v2f b0 = *(const v2f*)&bt[boff0 + kk];
            v2f b1 = *(const v2f*)&bt[boff1 + kk];
            c00 = __builtin_amdgcn_wmma_f32_16x16x4_f32(false, a0, false, b0, (short)0, c00, false, false);
            c01 = __builtin_amdgcn_wmma_f32_16x16x4_f32(false, a0, false, b1, (short)0, c01, false, false);
            c10 = __builtin_amdgcn_wmma_f32_16x16x4_f32(false, a1, false, b0, (short)0, c10, false, false);
            c11 = __builtin_amdgcn_wmma_f32_16x16x4_f32(false, a1, false, b1, (short)0, c11, false, false);
        }
        __syncthreads();
    }
#endif

    const int colA = n0 + wnw * 32 + r16;
    const int colB = colA + 16;
    #pragma unroll
    for (int at = 0; at < 2; ++at) {
        const v8f& ca = at ? c10 : c00;
        const v8f& cb = at ? c11 : c01;
        #pragma unroll
        for (int r = 0; r < 8; ++r) {
            int row = m0 + wmw * 32 + at * 16 + kh * 8 + r;
            if (row >= M) continue;
            float v0 = ca[r], v1 = cb[r];
            if (EPI == 0) {
                float s = dinv[row];
                v0 *= s; v1 *= s;
            } else {
                v0 += bias[colA];
                v1 += bias[colB];
                if (EPI == 1) { v0 = fmaxf(v0, 0.0f); v1 = fmaxf(v1, 0.0f); }
            }
            if (NCOUT == NC || colA < NCOUT) C[(size_t)row * NCOUT + colA] = v0;
            if (NCOUT == NC || colB < NCOUT) C[(size_t)row * NCOUT + colB] = v1;
        }
    }
}

// ---------------------------------------------------------------- aggregation
__global__ void k_zero4(float* __restrict__ S, long long n4) {
    long long i = (long long)blockIdx.x * 256 + threadIdx.x;
    if (i < n4) { float4 z{0.f, 0.f, 0.f, 0.f}; ((float4*)S)[i] = z; }
}

__global__ __launch_bounds__(256) void k_scatter(
    const long long* __restrict__ ei,
    const float* __restrict__ G, float* __restrict__ S)
{
    int e    = (blockIdx.x * 256 + threadIdx.x) >> 5;
    int lane = threadIdx.x & 31;
    if (e >= NE) return;
    int src = (int)ei[e];
    int dst = (int)ei[NE + e];
    const float* g = G + (size_t)src * HID;
    float*       s = S + (size_t)dst * HID;
    #pragma unroll
    for (int j = 0; j < 8; ++j) {
        int c = lane + 32 * j;
        atomicAdd(&s[c], g[c]);
    }
}

__global__ void k_finalize(const float* __restrict__ S, const float* __restrict__ G,
                           const float* __restrict__ dinv, const float* __restrict__ b,
                           float* __restrict__ Y)
{
    long long idx = (long long)blockIdx.x * 256 + threadIdx.x;
    if (idx >= (long long)NN * HID) return;
    int i = (int)(idx >> 8);
    int c = (int)(idx & 255);
    float v = dinv[i] * (S[idx] + G[idx]) + b[c];
    Y[idx] = fmaxf(v, 0.0f);
}

// ---------------------------------------------------------------- launch
extern "C" void kernel_launch(void* const* d_in, const int* in_sizes, int n_in,
                              void* d_out, int out_size, void* d_ws, size_t ws_size,
                              hipStream_t stream)
{
    const float*     x   = (const float*)d_in[0];
    const long long* ei  = (const long long*)d_in[1];
    const float* W1  = (const float*)d_in[2];  const float* b1  = (const float*)d_in[3];
    const float* W2  = (const float*)d_in[4];  const float* b2  = (const float*)d_in[5];
    const float* W3  = (const float*)d_in[6];  const float* b3  = (const float*)d_in[7];
    const float* Wf1 = (const float*)d_in[8];  const float* bf1 = (const float*)d_in[9];
    const float* Wf2 = (const float*)d_in[10]; const float* bf2 = (const float*)d_in[11];
    const float* Wf3 = (const float*)d_in[12]; const float* bf3 = (const float*)d_in[13];

    const size_t actElems = (size_t)NN * HID;
    float* P    = (float*)d_ws;                     // activations
    float* Q    = P + actElems;                     // gemm output g
    float* S    = Q + actElems;                     // scatter accumulator
    float* dinv = S + actElems;                     // rsqrt(deg)
    float* Xp   = dinv + NN;                        // padded input [NN,64]
    float* W1p  = Xp + (size_t)NN * FINP;           // padded W1    [64,256]
    float* Wfp  = W1p + FINP * HID;                 // padded Wf3   [256,128]
    float* bfp  = Wfp + HID * NCLSP;                // padded bf3   [128]

    const int TB = 256;
    const dim3 gN((NN + 63) / 64, HID / 128);       // 64x128 tiles
    const dim3 gC((NN + 63) / 64, NCLSP / 128);
    const long long n4   = (long long)actElems / 4;
    const int blkNodes   = (NN + TB - 1) / TB;
    const int blkEdges   = (NE + TB - 1) / TB;
    const int blkScatter = (int)(((long long)NE * 32 + TB - 1) / TB);
    const int blkElems   = (int)((actElems + TB - 1) / TB);
    const int blkZero    = (int)((n4 + TB - 1) / TB);

    // ---- preprocessing
    k_init_deg<<<blkNodes, TB, 0, stream>>>(dinv);
    k_deg_edges<<<blkEdges, TB, 0, stream>>>(ei, dinv);
    k_rsqrt_inplace<<<blkNodes, TB, 0, stream>>>(dinv);
    k_pad_x<<<(NN * FINP + TB - 1) / TB, TB, 0, stream>>>(x, Xp);
    k_pad_mat<<<(FINP * HID + TB - 1) / TB, TB, 0, stream>>>(W1, W1p, FIN, HID, FINP, HID);
    k_pad_mat<<<(HID * NCLSP + TB - 1) / TB, TB, 0, stream>>>(Wf3, Wfp, HID, NCLS, HID, NCLSP);
    k_pad_mat<<<1, TB, 0, stream>>>(bf3, bfp, 1, NCLS, 1, NCLSP);

    // ---- GCN layer 1 (padded K = 64)
    k_gemm<0, FINP, HID, HID><<<gN, TB, 0, stream>>>(Xp, W1p, Q, NN, dinv, nullptr);
    k_zero4<<<blkZero, TB, 0, stream>>>(S, n4);
    k_scatter<<<blkScatter, TB, 0, stream>>>(ei, Q, S);
    k_finalize<<<blkElems, TB, 0, stream>>>(S, Q, dinv, b1, P);

    // ---- GCN layer 2
    k_gemm<0, HID, HID, HID><<<gN, TB, 0, stream>>>(P, W2, Q, NN, dinv, nullptr);
    k_zero4<<<blkZero, TB, 0, stream>>>(S, n4);
    k_scatter<<<blkScatter, TB, 0, stream>>>(ei, Q, S);
    k_finalize<<<blkElems, TB, 0, stream>>>(S, Q, dinv, b2, P);

    // ---- GCN layer 3
    k_gemm<0, HID, HID, HID><<<gN, TB, 0, stream>>>(P, W3, Q, NN, dinv, nullptr);
    k_zero4<<<blkZero, TB, 0, stream>>>(S, n4);
    k_scatter<<<blkScatter, TB, 0, stream>>>(ei, Q, S);
    k_finalize<<<blkElems, TB, 0, stream>>>(S, Q, dinv, b3, P);

    // ---- dense head
    k_gemm<1, HID, HID, HID><<<gN, TB, 0, stream>>>(P, Wf1, Q, NN, nullptr, bf1);
    k_gemm<1, HID, HID, HID><<<gN, TB, 0, stream>>>(Q, Wf2, P, NN, nullptr, bf2);
    k_gemm<2, HID, NCLSP, NCLS><<<gC, TB, 0, stream>>>(P, Wfp, (float*)d_out, NN, nullptr, bfp);

    (void)in_sizes; (void)n_in; (void)out_size; (void)ws_size;
}